// harmolearn_73469710565980
// MI455X (gfx1250) — compile-verified
//
#include <hip/hip_runtime.h>
#include <stdint.h>

#define CH   1024
#define T    8192
#define WIN  32
#define NS   (T - WIN)          /* 8160 scan steps, = 32*255 */
#define SC   255                /* time-chunk per winmax wave */
#define BETA 0.95f
#define THR  1.0f
#define WMINV -1.0f
#define WMAXV 3.0f

typedef float v2f  __attribute__((ext_vector_type(2)));
typedef float v8f  __attribute__((ext_vector_type(8)));
typedef unsigned int u32x4 __attribute__((ext_vector_type(4)));
typedef int  i32x8 __attribute__((ext_vector_type(8)));
typedef int  i32x4 __attribute__((ext_vector_type(4)));

// Order-preserving float<->uint encoding so unsigned atomicMax == float max.
__device__ __forceinline__ unsigned enc_f32(float f) {
  unsigned u = __float_as_uint(f);
  return (u & 0x80000000u) ? ~u : (u | 0x80000000u);
}
__device__ __forceinline__ float dec_f32(unsigned k) {
  return __uint_as_float((k & 0x80000000u) ? (k & 0x7FFFFFFFu) : ~k);
}

// ---------------- Kernel 0: init encoded-max buffer -----------------------
__global__ void init_mkeys(unsigned* __restrict__ mk) {
  int i = blockIdx.x * blockDim.x + threadIdx.x;
  if (i < NS) mk[i] = 0u;  // minimal key
}

// ---------------- Kernel 1: colsum = ones(1x1024) * inp via f32 WMMA ------
// Each wave produces 16 timestep sums: C(16x16) += ones(16x4) * inp(4x16),
// accumulated over 256 K-steps (1024 channels / 4). All C rows identical.
__global__ void colsum_wmma(const float* __restrict__ inp,
                            float* __restrict__ colsum) {
  int wave = threadIdx.x >> 5;
  int lane = threadIdx.x & 31;
  int n0   = (blockIdx.x * (blockDim.x >> 5) + wave) * 16;
  int col  = n0 + (lane & 15);
  int half = lane >> 4;  // 0: K rows {k0,k0+1}; 1: K rows {k0+2,k0+3}
  v2f a; a.x = 1.0f; a.y = 1.0f;          // A = all ones (layout-proof)
  v8f c = {0.f,0.f,0.f,0.f,0.f,0.f,0.f,0.f};
  const float* p = inp + (size_t)(half * 2) * T + col;
  for (int k0 = 0; k0 < CH; k0 += 4) {
    v2f b;
    b.x = p[0];
    b.y = p[T];
    p += (size_t)4 * T;
    c = __builtin_amdgcn_wmma_f32_16x16x4_f32(false, a, false, b,
                                              (short)0, c, false, false);
  }
  if (lane < 16) colsum[col] = c[0];  // row M=0, N=lane
}

// ---------------- Kernel 2: M[s] = max_ch window-sum, time-parallel -------
// grid (NS/SC time-chunks, CH/32 channel groups), block = one wave32.
__global__ void winmax(const float* __restrict__ inp,
                       const float* __restrict__ colsum,
                       unsigned* __restrict__ mkeys) {
  int lane = threadIdx.x;                 // 0..31
  int ch   = blockIdx.y * 32 + lane;
  int s0   = blockIdx.x * SC;
  const float* row = inp + (size_t)ch * T;
  float win = 0.f;
#pragma unroll
  for (int u = 0; u < WIN; ++u) win += row[s0 + u] * colsum[s0 + u];
  for (int s = s0; s < s0 + SC; ++s) {
    float m = win;
#pragma unroll
    for (int off = 16; off >= 1; off >>= 1)
      m = fmaxf(m, __shfl_xor(m, off, 32));
    if (lane == 0) atomicMax(&mkeys[s], enc_f32(m));
    win += row[s + WIN] * colsum[s + WIN] - row[s] * colsum[s];
  }
}

// ---------------- Kernel 3: per-channel sequential scan -------------------
// One thread per channel; colsum + M staged to LDS by the Tensor Data Mover.
__global__ void scan_leaky(const float* __restrict__ inp,
                           const unsigned* __restrict__ ws,  // [colsum f32 x T][mkeys u32 x NS]
                           float* __restrict__ out) {
  __shared__ float lds[T + NS];           // 65408 B
  int lane = threadIdx.x;                 // 0..31 (one wave per block)
  int ch   = blockIdx.x * 32 + lane;

  // --- TDM: contiguous 1-D copy of (T+NS) dwords from ws -> LDS ---
  uint64_t gaddr    = (uint64_t)(uintptr_t)ws;
  unsigned lds_base = (unsigned)(uintptr_t)lds;   // low 32 bits = LDS byte offset
  const unsigned NW = (unsigned)(T + NS);         // 16352 dwords
  u32x4 g0;
  g0.x = 1u;                                       // count=1, user mode
  g0.y = lds_base;                                 // lds_addr
  g0.z = (unsigned)(gaddr & 0xFFFFFFFFu);          // global_addr[31:0]
  g0.w = (unsigned)((gaddr >> 32) & 0x1FFFFFFu) | (2u << 30);  // addr[56:32] | type=2
  i32x8 g1;
  g1[0] = (int)(2u << 16);                         // data_size = 4B
  g1[1] = (int)((NW & 0xFFFFu) << 16);             // tensor_dim0[15:0]
  g1[2] = (int)((NW >> 16) & 0xFFFFu) | (1 << 16); // tensor_dim0[31:16] | tensor_dim1=1
  g1[3] = (int)((NW & 0xFFFFu) << 16);             // tensor_dim1 hi | tile_dim0
  g1[4] = 1;                                       // tile_dim1 = 1
  g1[5] = (int)NW;                                 // tensor_dim0_stride lo
  g1[6] = 0;
  g1[7] = 0;
  i32x4 z4 = {0, 0, 0, 0};
  i32x8 z8 = {0, 0, 0, 0, 0, 0, 0, 0};
  __builtin_amdgcn_tensor_load_to_lds(g0, g1, z4, z4, z8, 0);
  __builtin_amdgcn_s_wait_tensorcnt(0);
  __syncthreads();

  // Decode encoded max keys -> 1/(1e-5 + M) in place (off the serial path).
  for (int k = lane; k < NS; k += 32) {
    unsigned key = __float_as_uint(lds[T + k]);
    lds[T + k]   = 1.0f / (1e-5f + dec_f32(key));
  }
  __syncthreads();

  const float* row  = inp + (size_t)ch * T;
  float*       orow = out + (size_t)ch * T;

  // Pass-through head: out[:, :WIN] = inp[:, :WIN]
#pragma unroll
  for (int k = 0; k < WIN; k += 4)
    *(float4*)(orow + k) = *(const float4*)(row + k);

  // Register-resident 32-deep product window (fully unrolled -> VGPRs).
  float buf[WIN];
#pragma unroll
  for (int k = 0; k < WIN; ++k) buf[k] = row[k] * lds[k];

  float w = 1.0f, mem = 0.0f;
  float4 x4, spk4;
  for (int base = 0; base < NS; base += WIN) {
    float win = 0.f;                      // fresh sum each 32 steps: no drift
#pragma unroll
    for (int k = 0; k < WIN; ++k) win += buf[k];
#pragma unroll
    for (int j = 0; j < WIN; ++j) {
      int s = base + j;
      if ((j & 3) == 0) x4 = *(const float4*)(row + s + WIN);
      float x = ((j & 3) == 0) ? x4.x : ((j & 3) == 1) ? x4.y
              : ((j & 3) == 2) ? x4.z : x4.w;
      float cc = win * lds[T + s];        // win / (1e-5 + M[s])
      w = 0.5f * w + 0.5f * cc;
      float wc    = fminf(fmaxf(w, WMINV), WMAXV);
      float reset = (mem > THR) ? 1.0f : 0.0f;
      mem = BETA * mem + wc * x - reset * THR;
      float spk = (mem > THR) ? 1.0f : 0.0f;
      if      ((j & 3) == 0) spk4.x = spk;
      else if ((j & 3) == 1) spk4.y = spk;
      else if ((j & 3) == 2) spk4.z = spk;
      else                   spk4.w = spk;
      if ((j & 3) == 3) *(float4*)(orow + WIN + s - 3) = spk4;
      float pn = x * lds[s + WIN];        // new product enters window
      win += pn - buf[j];
      buf[j] = pn;
    }
  }
}

extern "C" void kernel_launch(void* const* d_in, const int* in_sizes, int n_in,
                              void* d_out, int out_size, void* d_ws, size_t ws_size,
                              hipStream_t stream) {
  (void)in_sizes; (void)n_in; (void)out_size; (void)ws_size;
  const float* inp    = (const float*)d_in[0];
  float*       out    = (float*)d_out;
  float*       colsum = (float*)d_ws;                 // T floats
  unsigned*    mkeys  = (unsigned*)d_ws + T;          // NS uints (contiguous!)

  init_mkeys<<<(NS + 255) / 256, 256, 0, stream>>>(mkeys);
  colsum_wmma<<<T / 16 / 8, 256, 0, stream>>>(inp, colsum);   // 64 blocks x 8 waves
  dim3 gB(NS / SC, CH / 32);                                  // (32, 32)
  winmax<<<gB, 32, 0, stream>>>(inp, colsum, mkeys);
  scan_leaky<<<CH / 32, 32, 0, stream>>>(inp, (const unsigned*)d_ws, out);
}